// WSDDNOutputLayers_55722905698378
// MI455X (gfx1250) — compile-verified
//
#include <hip/hip_runtime.h>

// ---------------------------------------------------------------------------
// WSDDN output layers for MI455X (gfx1250, wave32, WMMA).
// Pipeline: fp32 WMMA GEMM (cls/det logits) -> softmax product scores ->
// clipped boxes -> per-class greedy NMS (sorted, thresholded) -> top-100.
// ---------------------------------------------------------------------------

#define R_N   4096
#define D_N   4096
#define NC    21      // C+1
#define CC    20      // C
#define IMG_W 1216.0f
#define IMG_H 800.0f
#define SCORE_T 0.05f
#define NMS_T   0.5f
#define TOPK  100

typedef float v2f __attribute__((ext_vector_type(2)));
typedef float v8f __attribute__((ext_vector_type(8)));

// ---------------------------------------------------------------------------
// Kernel 1: logits GEMM via V_WMMA_F32_16X16X4_F32.
// One wave computes a 16-row tile of both cls and det logits (21 cols padded
// to two 16-wide N tiles). A-frag: lane<16 holds K=k..k+1, lane>=16 K=k+2..k+3
// (contiguous float2 per lane). B-frag mirrors that layout over N lanes.
// ---------------------------------------------------------------------------
__global__ void __launch_bounds__(128)
gemm_logits_kernel(const float* __restrict__ x,
                   const float* __restrict__ Wc, const float* __restrict__ bc,
                   const float* __restrict__ Wd, const float* __restrict__ bd,
                   float* __restrict__ cls, float* __restrict__ det)
{
    const int wave = threadIdx.x >> 5;
    const int lane = threadIdx.x & 31;
    const int tile = blockIdx.x * 4 + wave;      // 0..255 (grid=64, 4 waves)
    const int row0 = tile * 16;
    const int half = lane >> 4;                  // 0: K pair 0..1, 1: K pair 2..3
    const int l16  = lane & 15;

    const int n0 = l16;                          // N tile 0: cols 0..15
    const int n1 = 16 + l16;                     // N tile 1: cols 16..31 (valid <21)
    const bool ok1 = (n1 < NC);
    const int n1c = ok1 ? n1 : (NC - 1);

    v8f acc_c0 = {}, acc_c1 = {}, acc_d0 = {}, acc_d1 = {};

    const float* xrow = x + (size_t)(row0 + l16) * D_N;

    for (int k = 0; k < D_N; k += 4) {
        const int ka = k + 2 * half;
        __builtin_prefetch(xrow + ka + 128, 0, 1);   // global_prefetch_b8

        v2f a;
        a.x = xrow[ka + 0];
        a.y = xrow[ka + 1];

        const float* wc0 = Wc + (size_t)ka * NC;
        const float* wc1 = wc0 + NC;
        const float* wd0 = Wd + (size_t)ka * NC;
        const float* wd1 = wd0 + NC;

        v2f bc0, bc1, bd0, bd1;
        bc0.x = wc0[n0];  bc0.y = wc1[n0];
        bd0.x = wd0[n0];  bd0.y = wd1[n0];
        float tc0 = wc0[n1c], tc1 = wc1[n1c];
        float td0 = wd0[n1c], td1 = wd1[n1c];
        bc1.x = ok1 ? tc0 : 0.0f;  bc1.y = ok1 ? tc1 : 0.0f;
        bd1.x = ok1 ? td0 : 0.0f;  bd1.y = ok1 ? td1 : 0.0f;

        acc_c0 = __builtin_amdgcn_wmma_f32_16x16x4_f32(false, a, false, bc0,
                                                       (short)0, acc_c0, false, false);
        acc_c1 = __builtin_amdgcn_wmma_f32_16x16x4_f32(false, a, false, bc1,
                                                       (short)0, acc_c1, false, false);
        acc_d0 = __builtin_amdgcn_wmma_f32_16x16x4_f32(false, a, false, bd0,
                                                       (short)0, acc_d0, false, false);
        acc_d1 = __builtin_amdgcn_wmma_f32_16x16x4_f32(false, a, false, bd1,
                                                       (short)0, acc_d1, false, false);
    }

    // C/D layout: VGPR v, lanes 0-15 -> M=v, lanes 16-31 -> M=8+v; N = lane&15.
    const float bias_c0 = bc[n0];
    const float bias_d0 = bd[n0];
    const float bias_c1 = ok1 ? bc[n1c] : 0.0f;
    const float bias_d1 = ok1 ? bd[n1c] : 0.0f;
    for (int v = 0; v < 8; ++v) {
        const int m = half * 8 + v;
        const size_t r = (size_t)(row0 + m);
        cls[r * NC + n0] = acc_c0[v] + bias_c0;
        det[r * NC + n0] = acc_d0[v] + bias_d0;
        if (ok1) {
            cls[r * NC + n1] = acc_c1[v] + bias_c1;
            det[r * NC + n1] = acc_d1[v] + bias_d1;
        }
    }
}

// ---------------------------------------------------------------------------
// Kernel 2: per-column (axis=0) max & sum-of-exp for det softmax. 21 blocks.
// ---------------------------------------------------------------------------
__global__ void __launch_bounds__(256)
det_colstats_kernel(const float* __restrict__ det,
                    float* __restrict__ colmax, float* __restrict__ colsum)
{
    const int c = blockIdx.x;
    __shared__ float red[256];
    __shared__ float smax;
    const int tid = threadIdx.x;

    float m = -3.4e38f;
    for (int r = tid; r < R_N; r += 256) m = fmaxf(m, det[(size_t)r * NC + c]);
    red[tid] = m;
    __syncthreads();
    for (int s = 128; s > 0; s >>= 1) {
        if (tid < s) red[tid] = fmaxf(red[tid], red[tid + s]);
        __syncthreads();
    }
    if (tid == 0) smax = red[0];
    __syncthreads();

    float acc = 0.0f;
    for (int r = tid; r < R_N; r += 256) acc += __expf(det[(size_t)r * NC + c] - smax);
    red[tid] = acc;
    __syncthreads();
    for (int s = 128; s > 0; s >>= 1) {
        if (tid < s) red[tid] += red[tid + s];
        __syncthreads();
    }
    if (tid == 0) { colmax[c] = smax; colsum[c] = red[0]; }
}

// ---------------------------------------------------------------------------
// Kernel 3: scores = row_softmax(cls) * col_softmax(det), cols 0..19,
// plus clipped boxes. One thread per proposal row.
// ---------------------------------------------------------------------------
__global__ void __launch_bounds__(256)
scores_boxes_kernel(const float* __restrict__ cls, const float* __restrict__ det,
                    const float* __restrict__ colmax, const float* __restrict__ colsum,
                    const float* __restrict__ boxes,
                    float* __restrict__ scores, float* __restrict__ bclip)
{
    const int r = blockIdx.x * 256 + threadIdx.x;
    if (r >= R_N) return;

    float cl[NC];
    float m = -3.4e38f;
    for (int c = 0; c < NC; ++c) { cl[c] = cls[(size_t)r * NC + c]; m = fmaxf(m, cl[c]); }
    float s = 0.0f;
    for (int c = 0; c < NC; ++c) { cl[c] = __expf(cl[c] - m); s += cl[c]; }
    const float inv = 1.0f / s;

    for (int c = 0; c < CC; ++c) {
        const float det_sm = __expf(det[(size_t)r * NC + c] - colmax[c]) / colsum[c];
        scores[(size_t)r * CC + c] = cl[c] * inv * det_sm;
    }

    bclip[r * 4 + 0] = fminf(fmaxf(boxes[r * 4 + 0], 0.0f), IMG_W);
    bclip[r * 4 + 1] = fminf(fmaxf(boxes[r * 4 + 1], 0.0f), IMG_H);
    bclip[r * 4 + 2] = fminf(fmaxf(boxes[r * 4 + 2], 0.0f), IMG_W);
    bclip[r * 4 + 3] = fminf(fmaxf(boxes[r * 4 + 3], 0.0f), IMG_H);
}

// ---------------------------------------------------------------------------
// Kernel 4: per-class greedy NMS. One block per class (20 blocks, 1024 thr).
// Pack (score, ~idx) into u64 so bitonic-descending gives stable argsort(-sc).
// Greedy loop runs only over candidates with score > SCORE_T (sorted prefix).
// ---------------------------------------------------------------------------
#define NMS_NT 1024
__global__ void __launch_bounds__(NMS_NT)
nms_kernel(const float* __restrict__ scores, const float* __restrict__ bclip,
           float* __restrict__ kept)
{
    const int c = blockIdx.x;
    const int tid = threadIdx.x;

    __shared__ unsigned long long keys[R_N];   // 32 KB
    __shared__ float sarea[R_N];               // 16 KB
    __shared__ unsigned char ssupp[R_N];       // 4 KB
    __shared__ unsigned char skeep[R_N];       // 4 KB
    __shared__ int s_ncand;

    if (tid == 0) s_ncand = 0;
    for (int i = tid; i < R_N; i += NMS_NT) {
        const float sc = scores[(size_t)i * CC + c];
        keys[i] = ((unsigned long long)__float_as_uint(sc) << 32) |
                  (unsigned int)(~(unsigned int)i);
        const float4 bb = ((const float4*)bclip)[i];
        sarea[i] = (bb.z - bb.x) * (bb.w - bb.y);
        ssupp[i] = 0;
        skeep[i] = 0;
    }
    __syncthreads();

    // Bitonic sort, descending on packed key.
    for (int k = 2; k <= R_N; k <<= 1) {
        for (int j = k >> 1; j > 0; j >>= 1) {
            for (int i = tid; i < R_N; i += NMS_NT) {
                const int ixj = i ^ j;
                if (ixj > i) {
                    const unsigned long long a = keys[i];
                    const unsigned long long b = keys[ixj];
                    const bool do_swap = ((i & k) == 0) ? (a < b) : (a > b);
                    if (do_swap) { keys[i] = b; keys[ixj] = a; }
                }
            }
            __syncthreads();
        }
    }

    // Candidate count: scores > SCORE_T occupy a sorted prefix.
    for (int i = tid; i < R_N; i += NMS_NT) {
        const float sc = __uint_as_float((unsigned int)(keys[i] >> 32));
        if (sc > SCORE_T) atomicAdd(&s_ncand, 1);
    }
    __syncthreads();
    const int ncand = s_ncand;

    // Greedy suppression (uniform branch on shared state each step).
    for (int i = 0; i < ncand; ++i) {
        if (!ssupp[i]) {
            if (tid == 0) skeep[i] = 1;
            const int idx_i = (int)(~(unsigned int)keys[i]);
            const float4 bi = ((const float4*)bclip)[idx_i];
            const float  ai = sarea[idx_i];
            for (int j = i + 1 + tid; j < ncand; j += NMS_NT) {
                if (!ssupp[j]) {
                    const int idx_j = (int)(~(unsigned int)keys[j]);
                    const float4 bj = ((const float4*)bclip)[idx_j];
                    const float lx = fmaxf(bi.x, bj.x), ly = fmaxf(bi.y, bj.y);
                    const float rx = fminf(bi.z, bj.z), ry = fminf(bi.w, bj.w);
                    const float w = fmaxf(rx - lx, 0.0f), h = fmaxf(ry - ly, 0.0f);
                    const float inter = w * h;
                    const float iou = inter / (ai + sarea[idx_j] - inter + 1e-9f);
                    if (iou > NMS_T) ssupp[j] = 1;
                }
            }
        }
        __syncthreads();
    }

    // Full rewrite of kept column (replay-safe: no stale state survives).
    for (int i = tid; i < R_N; i += NMS_NT) {
        const unsigned long long k = keys[i];
        const int idx = (int)(~(unsigned int)k);
        const float sc = __uint_as_float((unsigned int)(k >> 32));
        kept[(size_t)idx * CC + c] = skeep[i] ? sc : 0.0f;
    }
}

// ---------------------------------------------------------------------------
// Kernel 5: top-100 over kept scores (flat 4096*20). Single block; 100 rounds
// of packed-u64 argmax (ties -> lowest flat index, matching lax.top_k).
// ---------------------------------------------------------------------------
__global__ void __launch_bounds__(1024)
topk_kernel(float* __restrict__ kept, const float* __restrict__ bclip,
            float* __restrict__ out)
{
    __shared__ unsigned long long red[1024];
    const int tid = threadIdx.x;
    const int NFLAT = R_N * CC;

    float* top_vals  = out;              // [100]
    float* top_boxes = out + TOPK;       // [100*4]
    float* top_cls   = out + TOPK * 5;   // [100]

    for (int t = 0; t < TOPK; ++t) {
        unsigned long long best = 0ull;
        for (int i = tid; i < NFLAT; i += 1024) {
            const float v = kept[i];
            if (v >= 0.0f) {
                const unsigned long long k =
                    ((unsigned long long)__float_as_uint(v) << 32) |
                    (unsigned int)(~(unsigned int)i);
                if (k > best) best = k;
            }
        }
        red[tid] = best;
        __syncthreads();
        for (int s = 512; s > 0; s >>= 1) {
            if (tid < s && red[tid + s] > red[tid]) red[tid] = red[tid + s];
            __syncthreads();
        }
        if (tid == 0) {
            const unsigned long long k = red[0];
            const int idx = (int)(~(unsigned int)k);
            const float v = __uint_as_float((unsigned int)(k >> 32));
            const int prop = idx / CC;
            const int cls  = idx - prop * CC;
            top_vals[t] = v;
            top_boxes[t * 4 + 0] = bclip[prop * 4 + 0];
            top_boxes[t * 4 + 1] = bclip[prop * 4 + 1];
            top_boxes[t * 4 + 2] = bclip[prop * 4 + 2];
            top_boxes[t * 4 + 3] = bclip[prop * 4 + 3];
            top_cls[t] = (float)cls;
            kept[idx] = -1.0f;           // exclude from later rounds
        }
        __syncthreads();
    }
}

// ---------------------------------------------------------------------------
extern "C" void kernel_launch(void* const* d_in, const int* in_sizes, int n_in,
                              void* d_out, int out_size, void* d_ws, size_t ws_size,
                              hipStream_t stream)
{
    (void)in_sizes; (void)n_in; (void)out_size; (void)ws_size;

    const float* x     = (const float*)d_in[0];   // R*D
    const float* boxes = (const float*)d_in[1];   // R*4
    const float* W_cls = (const float*)d_in[2];   // D*21
    const float* b_cls = (const float*)d_in[3];   // 21
    const float* W_det = (const float*)d_in[4];   // D*21
    const float* b_det = (const float*)d_in[5];   // 21
    float* out = (float*)d_out;                   // 600 floats

    float* ws = (float*)d_ws;
    float* cls    = ws;                    // 4096*21
    float* det    = cls    + R_N * NC;     // 4096*21
    float* colmax = det    + R_N * NC;     // 32 (pad)
    float* colsum = colmax + 32;           // 32 (pad)
    float* scores = colsum + 32;           // 4096*20
    float* kept   = scores + R_N * CC;     // 4096*20
    float* bclip  = kept   + R_N * CC;     // 4096*4

    gemm_logits_kernel<<<64, 128, 0, stream>>>(x, W_cls, b_cls, W_det, b_det, cls, det);
    det_colstats_kernel<<<NC, 256, 0, stream>>>(det, colmax, colsum);
    scores_boxes_kernel<<<R_N / 256, 256, 0, stream>>>(cls, det, colmax, colsum,
                                                       boxes, scores, bclip);
    nms_kernel<<<CC, NMS_NT, 0, stream>>>(scores, bclip, kept);
    topk_kernel<<<1, 1024, 0, stream>>>(kept, bclip, out);
}